// OneHopGraphConvolution_58720792871766
// MI455X (gfx1250) — compile-verified
//
#include <hip/hip_runtime.h>

typedef __attribute__((ext_vector_type(2))) float v2f;
typedef __attribute__((ext_vector_type(8))) float v8f;

#define D_FEAT 128

// ---------------------------------------------------------------------------
// Phase 0: zero the scatter accumulator (d_ws is poisoned / stale each call)
// ---------------------------------------------------------------------------
__global__ void ohgc_zero_kernel(float4* __restrict__ p, int n4) {
    int i = blockIdx.x * blockDim.x + threadIdx.x;
    if (i < n4) p[i] = make_float4(0.f, 0.f, 0.f, 0.f);
}

// ---------------------------------------------------------------------------
// Phase 1: per-edge gather + scale + scatter-add.
// One wave32 per edge: lane l loads float4 of sender features [4l..4l+3]
// (32 lanes x 4 = 128 = D_FEAT) and atomically accumulates into the
// receiver row. x (25.6 MB) and the accumulator live in the 192 MB L2, so
// this is L2-atomic-throughput bound; f32 hardware atomics via relaxed
// agent-scope fetch_add -> global_atomic_add_f32.
// ---------------------------------------------------------------------------
__global__ void __launch_bounds__(256)
ohgc_edge_scatter_kernel(const float* __restrict__ x,
                         const float* __restrict__ edge_attr,
                         const int*   __restrict__ edge_index,
                         float*       __restrict__ conv,
                         int nEdges) {
    int gid  = blockIdx.x * blockDim.x + threadIdx.x;
    int edge = gid >> 5;
    int lane = gid & 31;
    if (edge >= nEdges) return;

    int   recv = edge_index[edge];            // row 0 = receivers
    int   send = edge_index[nEdges + edge];   // row 1 = senders
    float w    = edge_attr[edge];

    const float4* xs = (const float4*)(x + (size_t)send * D_FEAT);
    float4 v = xs[lane];

    float* dst = conv + (size_t)recv * D_FEAT + (size_t)lane * 4;
    __hip_atomic_fetch_add(dst + 0, v.x * w, __ATOMIC_RELAXED, __HIP_MEMORY_SCOPE_AGENT);
    __hip_atomic_fetch_add(dst + 1, v.y * w, __ATOMIC_RELAXED, __HIP_MEMORY_SCOPE_AGENT);
    __hip_atomic_fetch_add(dst + 2, v.z * w, __ATOMIC_RELAXED, __HIP_MEMORY_SCOPE_AGENT);
    __hip_atomic_fetch_add(dst + 3, v.w * w, __ATOMIC_RELAXED, __HIP_MEMORY_SCOPE_AGENT);
}

// ---------------------------------------------------------------------------
// Phase 2: out = conv @ W + b via V_WMMA_F32_16X16X4_F32 (full f32 WMMA).
// Block = 256 threads = 8 waves. Block b owns rows [16b, 16b+16); wave w owns
// cols [16w, 16w+16) -> one block emits a full 16x128 output stripe, A rows
// shared in L1 across the 8 waves. K loop: 128/4 = 32 wmma issues per wave.
//
// ISA VGPR layouts (CDNA5 ISA):
//  A 16x4 f32 (2 VGPRs): lanes 0-15 -> M=lane, {K0,K1}; lanes 16-31 -> {K2,K3}
//  B 4x16 f32 (2 VGPRs): VGPR0: lanes 0-15 row K0, lanes 16-31 row K2;
//                        VGPR1: rows K1 / K3 (rows striped across lanes)
//  C/D 16x16 f32 (8 VGPRs): VGPR r: lanes 0-15 -> M=r, lanes 16-31 -> M=r+8
// ---------------------------------------------------------------------------
__global__ void __launch_bounds__(256)
ohgc_gemm_wmma_kernel(const float* __restrict__ conv,
                      const float* __restrict__ Wm,
                      const float* __restrict__ bias,
                      float*       __restrict__ out) {
    int wave = threadIdx.x >> 5;     // 0..7 -> N-tile
    int lane = threadIdx.x & 31;
    int m0 = blockIdx.x * 16;
    int n0 = wave * 16;

    int l15   = lane & 15;
    int khalf = (lane >> 4) * 2;     // 0 for lanes 0-15, 2 for lanes 16-31

    const float* aBase = conv + (size_t)(m0 + l15) * D_FEAT + khalf;
    const float* bBase = Wm   + (size_t)khalf * D_FEAT + (n0 + l15);

    v8f c = {};
#pragma unroll
    for (int ko = 0; ko < D_FEAT; ko += 4) {
        v2f a;
        a.x = aBase[ko + 0];                      // A[m][ko+khalf]
        a.y = aBase[ko + 1];                      // A[m][ko+khalf+1]
        v2f bv;
        bv.x = bBase[(size_t)(ko + 0) * D_FEAT];  // W[ko+khalf  ][n]
        bv.y = bBase[(size_t)(ko + 1) * D_FEAT];  // W[ko+khalf+1][n]
        // 8 args: (neg_a, A, neg_b, B, c_mod, C, reuse_a, reuse_b)
        c = __builtin_amdgcn_wmma_f32_16x16x4_f32(
                false, a, false, bv, (short)0, c, false, false);
    }

    float bcol   = bias[n0 + l15];
    int   rowBase = m0 + ((lane >> 4) * 8);
#pragma unroll
    for (int r = 0; r < 8; ++r) {
        out[(size_t)(rowBase + r) * D_FEAT + (n0 + l15)] = c[r] + bcol;
    }
}

// Scalar tail for M % 16 != 0 (not hit for N_NODES = 50000, kept for safety;
// keeps the WMMA kernel free of partial tiles so EXEC stays all-ones).
__global__ void ohgc_gemm_tail_kernel(const float* __restrict__ conv,
                                      const float* __restrict__ Wm,
                                      const float* __restrict__ bias,
                                      float*       __restrict__ out,
                                      int rowStart, int nNodes) {
    int idx = blockIdx.x * blockDim.x + threadIdx.x;
    int row = rowStart + idx / D_FEAT;
    int col = idx % D_FEAT;
    if (row >= nNodes) return;
    float acc = bias[col];
    for (int k = 0; k < D_FEAT; ++k)
        acc += conv[(size_t)row * D_FEAT + k] * Wm[(size_t)k * D_FEAT + col];
    out[(size_t)row * D_FEAT + col] = acc;
}

extern "C" void kernel_launch(void* const* d_in, const int* in_sizes, int n_in,
                              void* d_out, int out_size, void* d_ws, size_t ws_size,
                              hipStream_t stream) {
    const float* x         = (const float*)d_in[0];   // [N, 128] f32
    const float* edge_attr = (const float*)d_in[1];   // [E, 1]  f32
    const float* Wm        = (const float*)d_in[2];   // [128, 128] f32
    const float* bias      = (const float*)d_in[3];   // [128] f32
    const int*   edge_index= (const int*)  d_in[4];   // [2, E] i32

    const int nNodes = in_sizes[0] / D_FEAT;          // 50000
    const int nEdges = in_sizes[1];                   // 600000

    float* conv = (float*)d_ws;                       // [N, 128] accumulator
    float* out  = (float*)d_out;

    // Phase 0: clear accumulator
    int n4 = (nNodes * D_FEAT) / 4;
    ohgc_zero_kernel<<<(n4 + 255) / 256, 256, 0, stream>>>((float4*)conv, n4);

    // Phase 1: edge gather/scale/scatter — one wave32 per edge
    long long threads = (long long)nEdges * 32;
    int blocks = (int)((threads + 255) / 256);
    ohgc_edge_scatter_kernel<<<blocks, 256, 0, stream>>>(x, edge_attr, edge_index,
                                                         conv, nEdges);

    // Phase 2: WMMA GEMM over full 16-row tiles
    int mTiles = nNodes / 16;
    if (mTiles > 0)
        ohgc_gemm_wmma_kernel<<<mTiles, 256, 0, stream>>>(conv, Wm, bias, out);

    // Scalar tail for leftover rows (none for 50000, but stay shape-generic)
    int tailRows = nNodes - mTiles * 16;
    if (tailRows > 0) {
        int tailElems = tailRows * D_FEAT;
        ohgc_gemm_tail_kernel<<<(tailElems + 255) / 256, 256, 0, stream>>>(
            conv, Wm, bias, out, mTiles * 16, nNodes);
    }
}